// ResidualBlock2_0_65377992180606
// MI455X (gfx1250) — compile-verified
//
#include <hip/hip_runtime.h>
#include <hip/hip_bf16.h>
#include <stdint.h>

#define NB 32
#define CH 32
#define HH 32
#define WW 32
#define HW (HH*WW)            // 1024
#define CHW (CH*HW)           // 32768
#define NCHW (NB*CHW)         // 1048576
#define WSZ (CH*CH*9)         // 9216

// workspace layout (float offsets)
#define OFF_WQ1  0
#define OFF_WQ2  9216
#define OFF_WB1  18432
#define OFF_WB2  18464
#define OFF_MU1  18496
#define OFF_RI1  18528
#define OFF_MU2  18560
#define OFF_RI2  18592
#define OFF_OUT1 32768
#define OFF_OUT2 (32768 + NCHW)

typedef float v2f __attribute__((ext_vector_type(2)));
typedef float v8f __attribute__((ext_vector_type(8)));

// ---- CDNA5 async global->LDS copy (ASYNCcnt-tracked) -----------------------
__device__ __forceinline__ void async_b128(void* lds, const void* gmem) {
    uint32_t l = (uint32_t)(uintptr_t)lds;   // low 32 bits of flat LDS addr == DS offset
    asm volatile("global_load_async_to_lds_b128 %0, %1, off"
                 :: "v"(l), "v"(gmem) : "memory");
}
__device__ __forceinline__ void wait_async0() {
    asm volatile("s_wait_asynccnt 0" ::: "memory");
}

// ---- kernel 1: quantize weights, transpose to [ci*9+k][co], compute wb -----
__global__ void k_quant(const float* __restrict__ w1, const float* __restrict__ w2,
                        float* __restrict__ ws) {
    const float scale = 2.5f / 127.f;
    const float inv   = 127.f / 2.5f;
    const int b = blockIdx.x;
    const float* w = b ? w2 : w1;
    float* wqT = ws + (b ? OFF_WQ2 : OFF_WQ1);
    float* wb  = ws + (b ? OFF_WB2 : OFF_WB1);
    const int co = threadIdx.x;            // 32 threads, one wave
    float s = 0.f;
    for (int j = 0; j < 288; ++j) {        // j = ci*9 + kh*3 + kw
        float v = w[co*288 + j] * inv;
        v = rintf(v);                      // round-half-even, matches jnp.round
        v = fminf(fmaxf(v, -127.f), 127.f);
        float q = v * scale;
        wqT[j*32 + co] = q;
        s += fabsf(q);
    }
    wb[co] = s * (1.f / 288.f);
}

// ---- kernel 2/4: adder conv over an 8-row tile, all channels in LDS --------
template<bool DO_BN>
__global__ void k_conv(const float* __restrict__ in, const float* __restrict__ wqT,
                       float* __restrict__ out,
                       const float* __restrict__ gamma, const float* __restrict__ beta,
                       const float* __restrict__ wb, const float* __restrict__ mu,
                       const float* __restrict__ rinv) {
    __shared__ float s_w[WSZ];             // [ci*9+k][co] -> bank-conflict-free over co
    __shared__ float s_x[CH][10][WW];      // rows r0-1 .. r0+8 (zero halo)
    __shared__ float s_scale[CH];
    __shared__ float s_bias[CH];

    const int tid = threadIdx.x;           // 256 threads
    const int n   = blockIdx.x >> 2;
    const int r0  = (blockIdx.x & 3) * 8;

    if (DO_BN && tid < CH) {
        float g = gamma[tid] * rinv[tid];
        s_scale[tid] = g;
        s_bias[tid]  = beta[tid] + wb[tid] - g * mu[tid];
    }

    // weights: 9216 floats = 2304 async b128 transfers
    #pragma unroll
    for (int i = 0; i < 9; ++i) {
        int idx = i*256 + tid;
        async_b128(&s_w[idx*4], wqT + idx*4);
    }
    // x tile: 32ch x 10rows x 32cols = 2560 b128 slots (invalid rows zero-filled)
    const float* xin = in + n*CHW;
    #pragma unroll
    for (int i = 0; i < 10; ++i) {
        int slot = i*256 + tid;
        int ci   = slot / 80;
        int rem  = slot % 80;
        int rr   = rem >> 3;
        int seg  = rem & 7;
        int r    = r0 - 1 + rr;
        float* dst = &s_x[ci][rr][seg*4];
        if (0 <= r && r < HH) {
            async_b128(dst, xin + ci*HW + r*WW + seg*4);
        } else {
            *(float4*)dst = make_float4(0.f, 0.f, 0.f, 0.f);
        }
    }
    wait_async0();
    __syncthreads();

    if (DO_BN) {                           // BN1 + ReLU applied in-place (valid rows only;
        for (int e = tid; e < CH*10*WW; e += 256) {   // halo zeros stay zeros = pad-after-bn)
            int ci  = e / (10*WW);
            int rem = e % (10*WW);
            int rr  = rem / WW;
            int col = rem % WW;
            int r   = r0 - 1 + rr;
            if (0 <= r && r < HH) {
                float v = s_x[ci][rr][col];
                s_x[ci][rr][col] = fmaxf(v * s_scale[ci] + s_bias[ci], 0.f);
            }
        }
        __syncthreads();
    }

    const int co   = tid & 31;             // lane -> output channel (broadcast x, strided w)
    const int row8 = tid >> 5;             // wave -> output row within tile

    float acc[WW];
    #pragma unroll
    for (int c = 0; c < WW; ++c) acc[c] = 0.f;

    for (int ci = 0; ci < CH; ++ci) {
        float wv[9];
        #pragma unroll
        for (int k = 0; k < 9; ++k) wv[k] = s_w[(ci*9 + k)*32 + co];
        #pragma unroll
        for (int kh = 0; kh < 3; ++kh) {
            const float* row = &s_x[ci][row8 + kh][0];
            float xm1 = 0.f;               // left zero pad
            float x0  = row[0];
            #pragma unroll
            for (int col = 0; col < WW; ++col) {
                float xp1 = (col < WW-1) ? row[col+1] : 0.f;   // right zero pad
                acc[col] += fabsf(xm1 - wv[kh*3+0])
                          + fabsf(x0  - wv[kh*3+1])
                          + fabsf(xp1 - wv[kh*3+2]);
                xm1 = x0; x0 = xp1;
            }
        }
    }
    float* op = out + ((n*CH + co)*HH + (r0 + row8))*WW;
    #pragma unroll
    for (int col = 0; col < WW; ++col) op[col] = -acc[col];
}

// ---- kernel 3/5: per-channel sum & sumsq via WMMA row-sum with ones --------
__device__ __forceinline__ v8f wmma_acc(v2f a, v2f b, v8f c) {
    // D = A(16x4 f32) x B(4x16 f32) + C ; B = ones -> every D column = row sums
    return __builtin_amdgcn_wmma_f32_16x16x4_f32(false, a, false, b,
                                                 (short)0, c, false, false);
}

__global__ void k_stats(const float* __restrict__ data,
                        float* __restrict__ mu, float* __restrict__ rinv) {
    const int c    = blockIdx.x;           // one channel per single-wave block
    const int lane = threadIdx.x;          // 32 threads, EXEC all ones
    v2f ones; ones.x = 1.f; ones.y = 1.f;
    v8f accs = {0.f,0.f,0.f,0.f,0.f,0.f,0.f,0.f};
    v8f accq = {0.f,0.f,0.f,0.f,0.f,0.f,0.f,0.f};
    const float* base0 = data + c*HW;
    for (int n = 0; n < NB; ++n) {
        const float* p = base0 + n*CHW;
        #pragma unroll 4
        for (int chunk = 0; chunk < 16; ++chunk) {   // 64 f32 per WMMA
            v2f a = *(const v2f*)(p + chunk*64 + lane*2);
            accs = wmma_acc(a, ones, accs);
            v2f a2; a2.x = a.x*a.x; a2.y = a.y*a.y;
            accq = wmma_acc(a2, ones, accq);
        }
    }
    // lane L holds rows M=0..7 (L<16) or M=8..15 (L>=16); total = lanes 0 + 16
    float ps = 0.f, pq = 0.f;
    #pragma unroll
    for (int i = 0; i < 8; ++i) { ps += accs[i]; pq += accq[i]; }
    float ts = __shfl(ps, 0, 32) + __shfl(ps, 16, 32);
    float tq = __shfl(pq, 0, 32) + __shfl(pq, 16, 32);
    if (lane == 0) {
        float m   = ts * (1.f / 32768.f);
        float var = tq * (1.f / 32768.f) - m * m;
        mu[c]   = m;
        rinv[c] = rsqrtf(var + 1e-5f);
    }
}

// ---- kernel 6: BN2 + residual + ReLU ---------------------------------------
__global__ void k_final(const float* __restrict__ conv, const float* __restrict__ resid,
                        const float* __restrict__ gamma, const float* __restrict__ beta,
                        const float* __restrict__ wb, const float* __restrict__ mu,
                        const float* __restrict__ rinv, float* __restrict__ out) {
    int idx = blockIdx.x * 256 + threadIdx.x;
    if (idx >= NCHW) return;
    int c = (idx >> 10) & 31;
    float g = gamma[c] * rinv[c];
    float v = conv[idx] * g + (beta[c] + wb[c] - g * mu[c]) + resid[idx];
    out[idx] = fmaxf(v, 0.f);
}

extern "C" void kernel_launch(void* const* d_in, const int* in_sizes, int n_in,
                              void* d_out, int out_size, void* d_ws, size_t ws_size,
                              hipStream_t stream) {
    const float* x      = (const float*)d_in[0];
    const float* w1     = (const float*)d_in[1];
    const float* gamma1 = (const float*)d_in[2];
    const float* beta1  = (const float*)d_in[3];
    const float* w2     = (const float*)d_in[4];
    const float* gamma2 = (const float*)d_in[5];
    const float* beta2  = (const float*)d_in[6];
    float* ws  = (float*)d_ws;
    float* out = (float*)d_out;

    float* wq1T = ws + OFF_WQ1;  float* wq2T = ws + OFF_WQ2;
    float* wb1  = ws + OFF_WB1;  float* wb2  = ws + OFF_WB2;
    float* mu1  = ws + OFF_MU1;  float* ri1  = ws + OFF_RI1;
    float* mu2  = ws + OFF_MU2;  float* ri2  = ws + OFF_RI2;
    float* out1 = ws + OFF_OUT1; float* out2 = ws + OFF_OUT2;

    k_quant<<<2, 32, 0, stream>>>(w1, w2, ws);

    k_conv<false><<<NB * 4, 256, 0, stream>>>(x, wq1T, out1,
                                              nullptr, nullptr, nullptr, nullptr, nullptr);
    k_stats<<<CH, 32, 0, stream>>>(out1, mu1, ri1);

    k_conv<true><<<NB * 4, 256, 0, stream>>>(out1, wq2T, out2,
                                             gamma1, beta1, wb1, mu1, ri1);
    k_stats<<<CH, 32, 0, stream>>>(out2, mu2, ri2);

    k_final<<<NCHW / 256, 256, 0, stream>>>(out2, x, gamma2, beta2, wb2, mu2, ri2, out);
}